// DeformableAttention_7378753814774
// MI455X (gfx1250) — compile-verified
//
#include <hip/hip_runtime.h>
#include <hip/hip_bf16.h>
#include <math.h>

// ---------------------------------------------------------------------------
// DeformableAttention on gfx1250 (MI455X), fp32 end-to-end.
//   K1: v = conv1x1(x, kv_w[C:2C])   WMMA f32 16x16x4; B staged via TDM->LDS
//   K2: offsets/mod 3x3 convs        VALU, LDS-tiled
//   K3: sampled tile in LDS -> out conv1x1 GEMM (WMMA f32, B from LDS)
// ---------------------------------------------------------------------------

#define Bn   8
#define Cn   256
#define Hn   128
#define Wn   128
#define HWn  (Hn * Wn)
#define HEADSn 8
#define HDn  32
#define GROUPSn 4

typedef float v2f __attribute__((ext_vector_type(2)));
typedef float v8f __attribute__((ext_vector_type(8)));

#ifndef __has_builtin
#define __has_builtin(x) 0
#endif
#if __has_builtin(__builtin_amdgcn_tensor_load_to_lds)
#define HAVE_TDM 1
#else
#define HAVE_TDM 0
#endif
#if __has_include(<hip/amd_detail/amd_gfx1250_TDM.h>)
#define TDM_ARGS6 1
#else
#define TDM_ARGS6 0
#endif

__device__ __forceinline__ v8f wmma_f32(v2f a, v2f b, v8f c) {
  // D = A(16x4) * B(4x16) + C, fp32, RNE
  return __builtin_amdgcn_wmma_f32_16x16x4_f32(
      /*neg_a=*/false, a, /*neg_b=*/false, b,
      /*c_mod=*/(short)0, c, /*reuse_a=*/false, /*reuse_b=*/false);
}

#if HAVE_TDM
typedef unsigned int uint32x4 __attribute__((ext_vector_type(4)));
typedef int int32x4 __attribute__((ext_vector_type(4)));
typedef int int32x8 __attribute__((ext_vector_type(8)));

// 2-D tile load Global->LDS via Tensor Data Mover (TENSORcnt tracked).
// D# packing per CDNA5 ISA.
// LDS pad: 8 DWORDs after every 64 DWORDs -> row stride 72 floats.
__device__ __forceinline__ void tdm_load_tile_f32(
    unsigned lds_byte, const float* gsrc,
    unsigned tile_w, unsigned tile_h, unsigned long long row_stride_elems) {
  unsigned long long ga = (unsigned long long)(const void*)gsrc;
  uint32x4 g0;
  g0.x = 1u;                                  // count=1, user-mode
  g0.y = lds_byte;                            // lds_addr (bytes)
  g0.z = (unsigned)ga;                        // global_addr[31:0]
  g0.w = (unsigned)((ga >> 32) & 0x1FFFFFFu)  // global_addr[56:32]
       | (2u << 30);                          // type = 2 ("image")
  const unsigned td0 = 1u << 20, td1 = 1u << 20;  // generous dims (tiles in-bounds)
  int32x8 g1;
  g1[0] = (int)((2u << 16)                    // data_size = 4 bytes
              | (1u << 20)                    // pad_enable
              | (5u << 22)                    // pad_interval: every 64 DWORDs
              | (7u << 25));                  // pad_amount: 8 DWORDs
  g1[1] = (int)((td0 & 0xFFFFu) << 16);                   // tensor_dim0 lo16
  g1[2] = (int)((td0 >> 16) | ((td1 & 0xFFFFu) << 16));   // dim0 hi | dim1 lo
  g1[3] = (int)((td1 >> 16) | ((tile_w & 0xFFFFu) << 16));// dim1 hi | tile_dim0
  g1[4] = (int)(tile_h & 0xFFFFu);                        // tile_dim1, tile_dim2=0
  g1[5] = (int)(unsigned)(row_stride_elems & 0xFFFFFFFFull); // dim0_stride lo32
  g1[6] = (int)(unsigned)(row_stride_elems >> 32);           // dim0_stride hi16
  g1[7] = 0;
  int32x4 z4 = {0, 0, 0, 0};
#if TDM_ARGS6
  int32x8 z8 = {0, 0, 0, 0, 0, 0, 0, 0};
  __builtin_amdgcn_tensor_load_to_lds(g0, g1, z4, z4, z8, 0);
#else
  __builtin_amdgcn_tensor_load_to_lds(g0, g1, z4, z4, 0);
#endif
}
#endif  // HAVE_TDM

// ---------------------------------------------------------------------------
// K1: Y[b][co][p] = bias[co] + sum_ci Wm[co][ci] * X[b][ci][p]
// block = 256 threads = 8 waves; tile M=128 (16/wave), N=64; K chunked by 32,
// X chunk staged in LDS by TDM (double-buffered), B fragments read from LDS.
// ---------------------------------------------------------------------------
#define KC   32
#define LDSW 72   // padded LDS row stride: rows kk / kk+2 hit disjoint banks

__global__ __launch_bounds__(256) void conv1x1_wmma_kernel(
    const float* __restrict__ X,    // [B][Cin][HW]
    const float* __restrict__ Wm,   // [Cout][Cin]
    const float* __restrict__ bias, // [Cout]
    float* __restrict__ Y,          // [B][Cout][HW]
    int Cin) {
  __shared__ float sx[2][KC][LDSW];

  const int lane  = threadIdx.x & 31;
  const int wave  = threadIdx.x >> 5;
  const int l15   = lane & 15;
  const int khalf = (lane >> 4) * 2;          // 0 or 2 (K-pair select)
  const int pbase = blockIdx.x * 64;
  const int mbase = blockIdx.y * 128 + wave * 16;
  const int b     = blockIdx.z;

  const float* Xtile = X + (size_t)b * Cin * HWn + pbase;  // [Cin][64] view
  float*       Yb    = Y + (size_t)b * Cn * HWn;
  const int NCH = Cin / KC;

#if HAVE_TDM
  if (threadIdx.x < 32)
    tdm_load_tile_f32((unsigned)(size_t)&sx[0][0][0], Xtile, 64, KC, HWn);
#else
  for (int i = threadIdx.x; i < KC * 64; i += 256)
    sx[0][i >> 6][i & 63] = Xtile[(size_t)(i >> 6) * HWn + (i & 63)];
#endif

  v8f acc[4] = {};

  for (int kc = 0; kc < NCH; ++kc) {
    const int buf = kc & 1;
#if HAVE_TDM
    if (threadIdx.x < 32) {
      if (kc + 1 < NCH) {
        tdm_load_tile_f32((unsigned)(size_t)&sx[buf ^ 1][0][0],
                          Xtile + (size_t)(kc + 1) * KC * HWn, 64, KC, HWn);
        __builtin_amdgcn_s_wait_tensorcnt(1);   // oldest (current chunk) done
      } else {
        __builtin_amdgcn_s_wait_tensorcnt(0);
      }
    }
#endif
    __syncthreads();

#pragma unroll
    for (int k4 = 0; k4 < KC; k4 += 4) {
      const int kk = k4 + khalf;
      const int kg = kc * KC + kk;
      v2f a;
      a.x = Wm[(size_t)(mbase + l15) * Cin + kg];
      a.y = Wm[(size_t)(mbase + l15) * Cin + kg + 1];
#pragma unroll
      for (int nt = 0; nt < 4; ++nt) {
        const int n = nt * 16 + l15;
        v2f bf;
        bf.x = sx[buf][kk][n];
        bf.y = sx[buf][kk + 1][n];
        acc[nt] = wmma_f32(a, bf, acc[nt]);
      }
    }
    __syncthreads();

#if !HAVE_TDM
    if (kc + 1 < NCH)
      for (int i = threadIdx.x; i < KC * 64; i += 256)
        sx[buf ^ 1][i >> 6][i & 63] =
            Xtile[(size_t)((kc + 1) * KC + (i >> 6)) * HWn + (i & 63)];
#endif
  }

#pragma unroll
  for (int nt = 0; nt < 4; ++nt) {
    const int n = pbase + nt * 16 + l15;
#pragma unroll
    for (int r = 0; r < 8; ++r) {
      const int m = mbase + r + (lane >> 4) * 8;   // C/D layout: hi lanes M+8
      Yb[(size_t)m * HWn + n] = acc[nt][r] + bias[m];
    }
  }
}

// ---------------------------------------------------------------------------
// K2: offsets (8ch) + mod (4ch) via 3x3 conv, LDS-tiled, 16x16 pixels/block.
// ---------------------------------------------------------------------------
#define CHK 8
__global__ __launch_bounds__(256) void offmod_kernel(
    const float* __restrict__ X,   // [B][C][HW]
    const float* __restrict__ OW,  // [8][C][3][3]
    const float* __restrict__ OB,  // [8]
    const float* __restrict__ MW,  // [4][C][3][3]
    const float* __restrict__ MB,  // [4]
    float* __restrict__ OFF,       // [B][8][HW]
    float* __restrict__ MOD) {     // [B][4][HW]
  __shared__ float sx[CHK][18][18];
  __shared__ float sw[CHK][12][9];

  const int b  = blockIdx.z;
  const int x0 = blockIdx.x * 16, y0 = blockIdx.y * 16;
  const int tid = threadIdx.x;
  const int px = tid & 15, py = tid >> 4;
  const float* Xb = X + (size_t)b * Cn * HWn;

  float acc[12];
#pragma unroll
  for (int i = 0; i < 12; ++i) acc[i] = 0.f;

  for (int c0 = 0; c0 < Cn; c0 += CHK) {
    for (int i = tid; i < CHK * 18 * 18; i += 256) {
      const int cc = i / 324, rem = i % 324;
      const int ty = rem / 18, tx = rem % 18;
      const int gy = y0 + ty - 1, gx = x0 + tx - 1;
      float v = 0.f;
      if (gy >= 0 && gy < Hn && gx >= 0 && gx < Wn)
        v = Xb[(size_t)(c0 + cc) * HWn + gy * Wn + gx];
      sx[cc][ty][tx] = v;
    }
    for (int i = tid; i < CHK * 12 * 9; i += 256) {
      const int cc = i / 108, rem = i % 108;
      const int oc = rem / 9, t = rem % 9;
      float wv;
      if (oc < 8) wv = OW[((size_t)oc * Cn + (c0 + cc)) * 9 + t];
      else        wv = MW[((size_t)(oc - 8) * Cn + (c0 + cc)) * 9 + t];
      sw[cc][oc][t] = wv;
    }
    __syncthreads();
#pragma unroll
    for (int cc = 0; cc < CHK; ++cc) {
#pragma unroll
      for (int ky = 0; ky < 3; ++ky) {
#pragma unroll
        for (int kx = 0; kx < 3; ++kx) {
          const float xv = sx[cc][py + ky][px + kx];
          const int t = ky * 3 + kx;
#pragma unroll
          for (int oc = 0; oc < 12; ++oc)
            acc[oc] = fmaf(xv, sw[cc][oc][t], acc[oc]);
        }
      }
    }
    __syncthreads();
  }

  const size_t p = (size_t)(y0 + py) * Wn + (x0 + px);
#pragma unroll
  for (int oc = 0; oc < 8; ++oc)
    OFF[((size_t)b * 8 + oc) * HWn + p] = acc[oc] + OB[oc];
#pragma unroll
  for (int g = 0; g < GROUPSn; ++g)
    MOD[((size_t)b * 4 + g) * HWn + p] = 1.0f + tanhf(acc[8 + g] + MB[g]);
}

// ---------------------------------------------------------------------------
// K3: build sampled[256][64] tile in LDS, then out = OW * sampled (WMMA).
// block = 256 threads = 8 waves; each wave: M=32 (2 tiles) x N=64 (4 tiles).
// ---------------------------------------------------------------------------
__global__ __launch_bounds__(256) void sample_outconv_kernel(
    const float* __restrict__ V,    // [B][256][HW]
    const float* __restrict__ OFF,  // [B][8][HW]
    const float* __restrict__ MOD,  // [B][4][HW]
    const float* __restrict__ OW,   // [256][256]
    const float* __restrict__ OB,   // [256]
    float* __restrict__ OUT) {      // [B][256][HW]
  __shared__ float s[256][64];      // sampled tile: [c_in][pixel] = 64 KB

  const int b     = blockIdx.y;
  const int pbase = blockIdx.x * 64;
  const int tid   = threadIdx.x;
  const float* Vb = V + (size_t)b * Cn * HWn;

  // ---- phase 1: bilinear sample + modulate, into LDS --------------------
  for (int t = tid; t < 64 * HEADSn; t += 256) {
    const int h  = t >> 6;          // head
    const int pl = t & 63;          // pixel within tile
    const int p  = pbase + pl;
    const int x  = p & (Wn - 1), y = p >> 7;
    const int g  = h & (GROUPSn - 1);

    const float ox = OFF[((size_t)b * 8 + 2 * g)     * HWn + p];
    const float oy = OFF[((size_t)b * 8 + 2 * g + 1) * HWn + p];
    const float mo = MOD[((size_t)b * 4 + g)         * HWn + p];

    // replicate reference normalize/denormalize exactly
    const float gx = 2.0f * ((float)x + ox) / (float)(Wn - 1) - 1.0f;
    const float gy = 2.0f * ((float)y + oy) / (float)(Hn - 1) - 1.0f;
    const float ix = ((gx + 1.0f) * (float)Wn - 1.0f) * 0.5f;
    const float iy = ((gy + 1.0f) * (float)Hn - 1.0f) * 0.5f;

    const float x0f = floorf(ix), y0f = floorf(iy);
    const float wx1 = ix - x0f, wx0 = 1.0f - wx1;
    const float wy1 = iy - y0f, wy0 = 1.0f - wy1;
    const int x0i = (int)x0f, y0i = (int)y0f;
    const int x1i = x0i + 1,  y1i = y0i + 1;
    const float vx0 = (x0i >= 0 && x0i < Wn) ? 1.f : 0.f;
    const float vx1 = (x1i >= 0 && x1i < Wn) ? 1.f : 0.f;
    const float vy0 = (y0i >= 0 && y0i < Hn) ? 1.f : 0.f;
    const float vy1 = (y1i >= 0 && y1i < Hn) ? 1.f : 0.f;
    const float w00 = wx0 * wy0 * vx0 * vy0 * mo;
    const float w01 = wx1 * wy0 * vx1 * vy0 * mo;
    const float w10 = wx0 * wy1 * vx0 * vy1 * mo;
    const float w11 = wx1 * wy1 * vx1 * vy1 * mo;
    const int cx0 = min(max(x0i, 0), Wn - 1), cx1 = min(max(x1i, 0), Wn - 1);
    const int cy0 = min(max(y0i, 0), Hn - 1), cy1 = min(max(y1i, 0), Hn - 1);
    const int i00 = cy0 * Wn + cx0, i01 = cy0 * Wn + cx1;
    const int i10 = cy1 * Wn + cx0, i11 = cy1 * Wn + cx1;

    const float* Vh = Vb + (size_t)h * HDn * HWn;
#pragma unroll 4
    for (int d = 0; d < HDn; ++d) {
      const float* Vc = Vh + (size_t)d * HWn;
      s[h * HDn + d][pl] =
          Vc[i00] * w00 + Vc[i01] * w01 + Vc[i10] * w10 + Vc[i11] * w11;
    }
  }
  __syncthreads();

  // ---- phase 2: GEMM out[256][64] = OW[256][256] * s[256][64] ------------
  const int lane  = tid & 31;
  const int wave  = tid >> 5;
  const int l15   = lane & 15;
  const int khalf = (lane >> 4) * 2;
  const int mb0   = wave * 32;

  v8f acc[2][4] = {};

  for (int k = 0; k < Cn; k += 4) {
    const int kk = k + khalf;
    v2f a0, a1;
    a0.x = OW[(size_t)(mb0 + l15) * Cn + kk];
    a0.y = OW[(size_t)(mb0 + l15) * Cn + kk + 1];
    a1.x = OW[(size_t)(mb0 + 16 + l15) * Cn + kk];
    a1.y = OW[(size_t)(mb0 + 16 + l15) * Cn + kk + 1];
#pragma unroll
    for (int nt = 0; nt < 4; ++nt) {
      const int n = nt * 16 + l15;
      v2f bf;
      bf.x = s[kk][n];
      bf.y = s[kk + 1][n];
      acc[0][nt] = wmma_f32(a0, bf, acc[0][nt]);
      acc[1][nt] = wmma_f32(a1, bf, acc[1][nt]);
    }
  }

  float* Ob = OUT + (size_t)b * Cn * HWn;
#pragma unroll
  for (int mt = 0; mt < 2; ++mt) {
#pragma unroll
    for (int nt = 0; nt < 4; ++nt) {
#pragma unroll
      for (int r = 0; r < 8; ++r) {
        const int m = mb0 + mt * 16 + r + (lane >> 4) * 8;
        const int n = pbase + nt * 16 + l15;
        Ob[(size_t)m * HWn + n] = acc[mt][nt][r] + OB[m];
      }
    }
  }
}

// ---------------------------------------------------------------------------
extern "C" void kernel_launch(void* const* d_in, const int* in_sizes, int n_in,
                              void* d_out, int out_size, void* d_ws, size_t ws_size,
                              hipStream_t stream) {
  const float* x        = (const float*)d_in[0];
  const float* offset_w = (const float*)d_in[1];
  const float* offset_b = (const float*)d_in[2];
  const float* mod_w    = (const float*)d_in[3];
  const float* mod_b    = (const float*)d_in[4];
  const float* kv_w     = (const float*)d_in[5];
  const float* kv_b     = (const float*)d_in[6];
  const float* out_w    = (const float*)d_in[7];
  const float* out_b    = (const float*)d_in[8];
  float* out = (float*)d_out;

  // workspace: v [B][C][HW] + offsets [B][8][HW] + mod [B][4][HW] (~140 MB)
  float* v   = (float*)d_ws;
  float* off = v + (size_t)Bn * Cn * HWn;
  float* mod = off + (size_t)Bn * 8 * HWn;

  // K1: v-projection (only the v half of kv is ever used)
  {
    dim3 grid(HWn / 64, Cn / 128, Bn);
    conv1x1_wmma_kernel<<<grid, 256, 0, stream>>>(
        x, kv_w + (size_t)Cn * Cn, kv_b + Cn, v, Cn);
  }
  // K2: offsets + modulators
  {
    dim3 grid(Wn / 16, Hn / 16, Bn);
    offmod_kernel<<<grid, 256, 0, stream>>>(x, offset_w, offset_b,
                                            mod_w, mod_b, off, mod);
  }
  // K3: fused bilinear sample + out-projection
  {
    dim3 grid(HWn / 64, Bn);
    sample_outconv_kernel<<<grid, 256, 0, stream>>>(v, off, mod,
                                                    out_w, out_b, out);
  }
}